// InvariantMessage_34376918237205
// MI455X (gfx1250) — compile-verified
//
#include <hip/hip_runtime.h>
#include <hip/hip_bf16.h>
#include <math.h>

#define N_ATOMS 50000
#define N_EDGES 800000
#define FEAT    128
#define N_RBF   20
#define CUTOFF  5.0f
#define PI_F    3.14159265358979323846f

typedef __attribute__((ext_vector_type(2))) float v2f;
typedef __attribute__((ext_vector_type(8))) float v8f;

// ---------------------------------------------------------------------------
// Kernel 1: inv = silu(s_j @ W1 + b1) @ W2 + b2
// One block = 16 atoms, 256 threads = 8 waves; wave w owns output columns
// [16w, 16w+16). K=128 reduced with 32 x V_WMMA_F32_16X16X4_F32.
// Intermediate h tile (16x128) lives in LDS between the two GEMMs.
// inv stores are regular-temporal: kernel 2 re-reads them out of L2.
// ---------------------------------------------------------------------------
__global__ __launch_bounds__(256)
void atom_mlp_kernel(const float* __restrict__ s_j,
                     const float* __restrict__ W1, const float* __restrict__ b1,
                     const float* __restrict__ W2, const float* __restrict__ b2,
                     float* __restrict__ inv_out) {
  __shared__ float hbuf[16][132];   // padded: stride 132 -> 16 lanes hit 16 banks

  const int tid   = threadIdx.x;
  const int lane  = tid & 31;
  const int wv    = tid >> 5;       // 0..7
  const int half  = lane >> 4;      // 0 | 1
  const int mr    = lane & 15;      // row (A/half) or col (B/C/D) index
  const int atom0 = blockIdx.x * 16;
  const int n0    = wv * 16;

  // ---- GEMM 1: h = silu(s_j @ W1 + b1) ----
  v8f acc = {};
  #pragma unroll 4
  for (int kb = 0; kb < FEAT; kb += 4) {
    const int k0 = kb + 2 * half;
    v2f a, b;
    // A 16x4 f32 layout: lanes0-15 hold K=kb+0/1, lanes16-31 hold K=kb+2/3
    a.x = s_j[(size_t)(atom0 + mr) * FEAT + k0];
    a.y = s_j[(size_t)(atom0 + mr) * FEAT + k0 + 1];
    // B 4x16: lane -> column n0+mr, same K pairing
    b.x = W1[(size_t)k0 * FEAT + n0 + mr];
    b.y = W1[(size_t)(k0 + 1) * FEAT + n0 + mr];
    acc = __builtin_amdgcn_wmma_f32_16x16x4_f32(false, a, false, b,
                                                (short)0, acc, false, false);
  }
  {
    const float bb = b1[n0 + mr];
    #pragma unroll
    for (int r = 0; r < 8; ++r) {
      // D layout: VGPR r -> row r (lanes0-15) / r+8 (lanes16-31), col = mr
      float x = acc[r] + bb;
      // silu via hardware v_rcp_f32 (avoids IEEE div-scale fixup chain)
      hbuf[r + 8 * half][n0 + mr] =
          x * __builtin_amdgcn_rcpf(1.0f + __expf(-x));
    }
  }
  __syncthreads();

  // ---- GEMM 2: inv = h @ W2 + b2 ----
  v8f acc2 = {};
  #pragma unroll 4
  for (int kb = 0; kb < FEAT; kb += 4) {
    const int k0 = kb + 2 * half;
    v2f a, b;
    a.x = hbuf[mr][k0];
    a.y = hbuf[mr][k0 + 1];
    b.x = W2[(size_t)k0 * FEAT + n0 + mr];
    b.y = W2[(size_t)(k0 + 1) * FEAT + n0 + mr];
    acc2 = __builtin_amdgcn_wmma_f32_16x16x4_f32(false, a, false, b,
                                                 (short)0, acc2, false, false);
  }
  {
    const float bb = b2[n0 + mr];
    #pragma unroll
    for (int r = 0; r < 8; ++r)
      inv_out[(size_t)(atom0 + r + 8 * half) * FEAT + n0 + mr] = acc2[r] + bb;
  }
}

// ---------------------------------------------------------------------------
// Kernel 2: per-edge  out = inv[nbrs[:,1]] * (rbf @ W_rbf + b_rbf) * env
// One block = 16 edges, 256 threads = 8 waves. RBF values (16x20) staged in
// LDS, then w = rbf @ W_rbf as 5 x V_WMMA_F32_16X16X4_F32 per wave, fused
// with envelope, the L2-resident gather (inv table = 25.6 MB << 192 MB L2)
// and a non-temporal 410 MB output stream so the write-once stores do not
// evict the gather table from L2.
// ---------------------------------------------------------------------------
__global__ __launch_bounds__(256)
void edge_kernel(const float* __restrict__ inv,
                 const float* __restrict__ dist,
                 const int*   __restrict__ nbrs,
                 const float* __restrict__ W_rbf,
                 const float* __restrict__ b_rbf,
                 float* __restrict__ out) {
  __shared__ float rbf[16][N_RBF];  // stride 20 -> 16 distinct banks, no pad
  __shared__ float envs[16];
  __shared__ int   jidx[16];

  const int tid = threadIdx.x;
  const int e0  = blockIdx.x * 16;

  for (int idx = tid; idx < 16 * N_RBF; idx += 256) {
    const int e = idx / N_RBF, r = idx % N_RBF;
    const float d = __builtin_nontemporal_load(&dist[e0 + e]);
    rbf[e][r] = __sinf((float)(r + 1) * (PI_F / CUTOFF) * d) / d;
  }
  if (tid < 16) {
    const float d = __builtin_nontemporal_load(&dist[e0 + tid]);
    envs[tid] = 0.5f * (__cosf(PI_F * d / CUTOFF) + 1.0f) * (d < CUTOFF ? 1.0f : 0.0f);
    jidx[tid] = __builtin_nontemporal_load(&nbrs[(size_t)(e0 + tid) * 2 + 1]);
  }
  __syncthreads();

  const int lane = tid & 31;
  const int wv   = tid >> 5;
  const int half = lane >> 4;
  const int mr   = lane & 15;
  const int n    = wv * 16 + mr;    // output channel

  v8f acc = {};
  #pragma unroll
  for (int kb = 0; kb < N_RBF; kb += 4) {
    const int k0 = kb + 2 * half;
    v2f a, b;
    a.x = rbf[mr][k0];
    a.y = rbf[mr][k0 + 1];
    b.x = W_rbf[(size_t)k0 * FEAT + n];
    b.y = W_rbf[(size_t)(k0 + 1) * FEAT + n];
    acc = __builtin_amdgcn_wmma_f32_16x16x4_f32(false, a, false, b,
                                                (short)0, acc, false, false);
  }

  const float bb = b_rbf[n];
  #pragma unroll
  for (int r = 0; r < 8; ++r) {
    const int e = r + 8 * half;
    const float w = (acc[r] + bb) * envs[e];
    const float v = w * inv[(size_t)jidx[e] * FEAT + n];
    __builtin_nontemporal_store(v, &out[(size_t)(e0 + e) * FEAT + n]);
  }
}

// ---------------------------------------------------------------------------
extern "C" void kernel_launch(void* const* d_in, const int* in_sizes, int n_in,
                              void* d_out, int out_size, void* d_ws, size_t ws_size,
                              hipStream_t stream) {
  const float* s_j   = (const float*)d_in[0];
  const float* dist  = (const float*)d_in[1];
  const int*   nbrs  = (const int*)  d_in[2];
  const float* W1    = (const float*)d_in[3];
  const float* b1    = (const float*)d_in[4];
  const float* W2    = (const float*)d_in[5];
  const float* b2    = (const float*)d_in[6];
  const float* W_rbf = (const float*)d_in[7];
  const float* b_rbf = (const float*)d_in[8];
  float* out = (float*)d_out;

  float* inv_ws = (float*)d_ws;     // 50000*128*4 = 25.6 MB scratch

  atom_mlp_kernel<<<N_ATOMS / 16, 256, 0, stream>>>(s_j, W1, b1, W2, b2, inv_ws);
  edge_kernel<<<N_EDGES / 16, 256, 0, stream>>>(inv_ws, dist, nbrs, W_rbf, b_rbf, out);
}